// DirectHODLR_18614388261383
// MI455X (gfx1250) — compile-verified
//
#include <hip/hip_runtime.h>

typedef __attribute__((ext_vector_type(2))) float v2f;
typedef __attribute__((ext_vector_type(8))) float v8f;

#define N_TOTAL 524288
#define DEPTH 8

// ranks per level (level 0 = coarsest, block = N/2)
__device__ const int d_rank[DEPTH]     = {64, 64, 64, 64, 32, 16, 8, 4};
// offsets into proj buffer: sum over prior levels of 2^(l+1)*r_l  (+ sentinel)
__device__ const int d_lvlOff[DEPTH+1] = {0, 128, 384, 896, 1920, 2944, 3968, 4992, 6016};
// offsets into per-block partial buffer: 256 chunks * r_l per level
__device__ const int d_pOff[DEPTH]     = {0, 16384, 32768, 49152, 65536, 73728, 77824, 79872};
// offsets of each level's proj vector inside the per-block LDS cache (+ sentinel)
__device__ const int d_plOff[DEPTH+1]  = {0, 64, 128, 192, 256, 288, 304, 312, 316};

struct UPtrs { const float* p[DEPTH]; };

// ---------------------------------------------------------------------------
// Kernel 1: per-(level, 2048-row chunk) partial projections  proj_k += u[i,k]*x[i]
// Thread layout t = k + r*j  ==>  u index = base*r + t + iter*256 : fully coalesced.
// ---------------------------------------------------------------------------
__global__ __launch_bounds__(256) void hodlr_proj_partial(UPtrs U,
                                                          const float* __restrict__ x,
                                                          float* __restrict__ partial)
{
    __shared__ float sdata[256];
    const int t     = threadIdx.x;
    const int level = blockIdx.y;
    const int chunk = blockIdx.x;
    const int base  = chunk * 2048;

    const int r      = d_rank[level];
    const int groups = 256 / r;          // r in {4..64} divides 256
    const int k      = t % r;
    const int j      = t / r;

    const float* __restrict__ u = U.p[level];

    float acc = 0.0f;
    for (int row = j; row < 2048; row += groups)
        acc += u[(base + row) * r + k] * x[base + row];

    sdata[t] = acc;
    __syncthreads();
    // deterministic LDS tree reduction over the j-groups for each rank k
    for (int s = groups >> 1; s > 0; s >>= 1) {
        if (j < s) sdata[k + r * j] += sdata[k + r * (j + s)];
        __syncthreads();
    }
    if (t < r)
        partial[d_pOff[level] + chunk * r + t] = sdata[t];
}

// ---------------------------------------------------------------------------
// Kernel 2: reduce per-chunk partials into proj[level][pair*2+child][k].
// One thread per proj element; child's chunks are contiguous (block >= 2048).
// ---------------------------------------------------------------------------
__global__ __launch_bounds__(256) void hodlr_proj_reduce(const float* __restrict__ partial,
                                                         float* __restrict__ proj)
{
    int g = blockIdx.x * blockDim.x + threadIdx.x;
    if (g >= 6016) return;
    int l = 0;
    while (l < DEPTH - 1 && g >= d_lvlOff[l + 1]) ++l;
    const int r    = d_rank[l];
    const int idx  = g - d_lvlOff[l];
    const int slot = idx / r;            // pair*2 + child
    const int k    = idx % r;
    const int bc   = 256 >> (l + 1);     // 2048-row chunks per child
    const int fst  = slot * bc;
    float s = 0.0f;
    for (int b = 0; b < bc; ++b)
        s += partial[d_pOff[l] + (fst + b) * r + k];
    proj[g] = s;
}

// ---------------------------------------------------------------------------
// Kernel 3: y = diag*x + sum_levels U * proj_swapped, via V_WMMA_F32_16X16X4_F32.
// 256 threads = 8 waves; each wave does two 16-row tiles. proj (sibling child)
// for all 8 levels is cached in LDS (316 floats); B operand broadcasts p to all
// 16 columns so every column of D holds the matvec result; accumulation across
// K-chunks and levels rides the WMMA C operand.
// ---------------------------------------------------------------------------
__global__ __launch_bounds__(256) void hodlr_update(UPtrs U,
                                                    const float* __restrict__ x,
                                                    const float* __restrict__ diag,
                                                    const float* __restrict__ proj,
                                                    float* __restrict__ out)
{
    __shared__ float psh[316];
    __shared__ float ysh[256];
    const int t    = threadIdx.x;
    const int base = blockIdx.x * 256;   // min block_size = 2048: whole block in one child

    // stage swapped-sibling projections for this row block (all levels) into LDS
    for (int q = t; q < 316; q += 256) {
        int l = 0;
        while (l < DEPTH - 1 && q >= d_plOff[l + 1]) ++l;
        const int k   = q - d_plOff[l];
        const int r   = d_rank[l];
        const int bs  = N_TOTAL >> (l + 1);
        const int c   = (base / bs) & 1;
        const int pr  = base / (2 * bs);
        const int sib = pr * 2 + (1 - c);
        psh[q] = proj[d_lvlOff[l] + sib * r + k];
    }
    __syncthreads();

    const int lane = t & 31;
    const int wave = t >> 5;
    const int m    = lane & 15;
    const int hi   = (lane >> 4) << 1;   // lanes 0-15 -> K{0,1}, lanes 16-31 -> K{2,3}

    for (int tile = 0; tile < 2; ++tile) {
        const int rowBase = base + wave * 32 + tile * 16;
        v8f acc = {0.f, 0.f, 0.f, 0.f, 0.f, 0.f, 0.f, 0.f};

        for (int l = 0; l < DEPTH; ++l) {
            const int r = d_rank[l];
            const float* __restrict__ u = U.p[l];
            const int rowOff = (rowBase + m) * r;
            const int po     = d_plOff[l];
            for (int kc = 0; kc < r; kc += 4) {
                v2f a, b;
                a.x = u[rowOff + kc + hi];
                a.y = u[rowOff + kc + hi + 1];
                b.x = psh[po + kc + hi];         // same column value for all N
                b.y = psh[po + kc + hi + 1];
                acc = __builtin_amdgcn_wmma_f32_16x16x4_f32(
                          false, a, false, b, (short)0, acc, false, false);
            }
        }
        // every column of D is identical: lane 0 -> rows 0..7, lane 16 -> rows 8..15
        if (lane == 0) {
            #pragma unroll
            for (int j2 = 0; j2 < 8; ++j2) ysh[wave * 32 + tile * 16 + j2] = acc[j2];
        }
        if (lane == 16) {
            #pragma unroll
            for (int j2 = 0; j2 < 8; ++j2) ysh[wave * 32 + tile * 16 + 8 + j2] = acc[j2];
        }
    }
    __syncthreads();

    const int i = base + t;               // coalesced final read-modify-write
    out[i] = diag[i] * x[i] + ysh[t];
}

// ---------------------------------------------------------------------------
// Host launcher. Workspace: proj = ws[0..6016), partials = ws[6016..86912).
// ---------------------------------------------------------------------------
extern "C" void kernel_launch(void* const* d_in, const int* in_sizes, int n_in,
                              void* d_out, int out_size, void* d_ws, size_t ws_size,
                              hipStream_t stream) {
    (void)in_sizes; (void)n_in; (void)out_size; (void)ws_size;

    const float* x    = (const float*)d_in[0];
    const float* diag = (const float*)d_in[1];
    UPtrs U;
    for (int i = 0; i < DEPTH; ++i) U.p[i] = (const float*)d_in[2 + i];

    float* proj    = (float*)d_ws;
    float* partial = proj + 6016;
    float* out     = (float*)d_out;

    dim3 gridP(256, DEPTH);
    hodlr_proj_partial<<<gridP, 256, 0, stream>>>(U, x, partial);
    hodlr_proj_reduce<<<(6016 + 255) / 256, 256, 0, stream>>>(partial, proj);
    hodlr_update<<<N_TOTAL / 256, 256, 0, stream>>>(U, x, diag, proj, out);
}